// QCNNHybridModel_65481071410252
// MI455X (gfx1250) — compile-verified
//
#include <hip/hip_runtime.h>

typedef __attribute__((ext_vector_type(2))) float v2f;
typedef __attribute__((ext_vector_type(8))) float v8f;

__device__ __forceinline__ v8f zero8() {
  v8f c;
#pragma unroll
  for (int i = 0; i < 8; ++i) c[i] = 0.0f;
  return c;
}

__device__ __forceinline__ float clip5(float w) { return fminf(fmaxf(w, -5.0f), 5.0f); }

__device__ __forceinline__ float fast_tanh(float x) {
#if __has_builtin(__builtin_amdgcn_tanhf)
  return __builtin_amdgcn_tanhf(x);   // gfx1250 v_tanh_f32 (TRANS)
#else
  float e = __expf(2.0f * x);         // NaN-safe fallback
  return 1.0f - 2.0f / (e + 1.0f);
#endif
}

__device__ __forceinline__ float fast_sigmoid(float x) {
  return 1.0f / (1.0f + __expf(-x));
}

// xor-16 half-wave exchange as a single-result VALU op (v_permlanex16_b32
// with identity per-lane selects). No LDS pipe, no dscnt waits.
__device__ __forceinline__ float permx16(float x) {
#if __has_builtin(__builtin_amdgcn_permlanex16)
  unsigned u = __builtin_bit_cast(unsigned, x);
  unsigned r =
      __builtin_amdgcn_permlanex16(u, u, 0x76543210u, 0xFEDCBA98u, false, false);
  return __builtin_bit_cast(float, r);
#else
  return __shfl_xor(x, 16, 32);
#endif
}

// D = A(16x4, f32) x B(4x16, f32) + C, full fp32 precision
__device__ __forceinline__ v8f wmma4(v2f a, v2f b, v8f c) {
  return __builtin_amdgcn_wmma_f32_16x16x4_f32(false, a, false, b, (short)0, c,
                                               false, false);
}

// A-fragment loader: W is (FOUT x FIN) row-major. A layout (32-bit 16x4):
// lane m=lane16 holds K = 4*j + 2*hi + v in frag[j][v]. Rows >= FOUT are zero.
template <int FOUT, int FIN, bool CLIP>
__device__ __forceinline__ void load_wfrag(const float* __restrict__ W, v2f* frag,
                                           int lane16, int hi) {
  constexpr int NS = (FIN + 3) / 4;
#pragma unroll
  for (int j = 0; j < NS; ++j) {
#pragma unroll
    for (int v = 0; v < 2; ++v) {
      const int k = 4 * j + 2 * hi + v;  // FIN is a multiple of 4 -> k < FIN
      float w = 0.0f;
      if (lane16 < FOUT) {
        w = W[lane16 * FIN + k];
        if (CLIP) w = clip5(w);
      }
      frag[j][v] = w;
    }
  }
}

// Per-lane selected (bias, scale, shift) for C layout: vgpr v holds feature
// m = v + 8*hi of this lane's batch row. Invalid features get 0 (their C is 0).
template <int FOUT, bool CLIPB>
__device__ __forceinline__ void load_bst(const float* __restrict__ b,
                                         const float* __restrict__ s,
                                         const float* __restrict__ t, float* bs,
                                         float* ss, float* ts, int hi) {
#pragma unroll
  for (int v = 0; v < 8; ++v) {
    const int m = v + 8 * hi;
    const bool ok = (m < FOUT);
    float bb = ok ? b[m] : 0.0f;
    if (CLIPB) bb = clip5(bb);
    bs[v] = bb;
    ss[v] = ok ? s[m] : 0.0f;
    ts[v] = ok ? t[m] : 0.0f;
  }
}

template <int FOUT>
__device__ __forceinline__ void activate(v8f c, const float* bs, const float* ss,
                                         const float* ts, float* act) {
  constexpr int VMAX = FOUT < 8 ? FOUT : 8;
#pragma unroll
  for (int v = 0; v < VMAX; ++v)
    act[v] = fast_tanh(c[v] + bs[v]) * ss[v] + ts[v];
#pragma unroll
  for (int v = VMAX; v < 8; ++v) act[v] = 0.0f;  // never read; keeps it defined
}

// ---------------------------------------------------------------------------
// C-layout -> next-layer B-fragment conversion.
// B element (slice j, vgpr v) needs feature k = 4*j + 2*hi + v of this lane's
// batch row; feature k lives in VGPR (k&7) of the lane in half k>>3.
// Exactly one of (lower-dest, upper-dest) sources crosses the half boundary,
// fetched with a single-result v_permlanex16, then merged by half-select.
// ---------------------------------------------------------------------------
template <int KTOT>
__device__ __forceinline__ void make_bfrag(const float* act, v2f* frag, int hi) {
#pragma unroll
  for (int j = 0; j < KTOT / 4; ++j) {
#pragma unroll
    for (int v = 0; v < 2; ++v) {
      const int kl = 4 * j + v;   // feature needed by lower-half dest lanes
      const int ku = kl + 2;      // feature needed by upper-half dest lanes
      float al = (kl < 8) ? act[kl] : permx16(act[kl & 7]);
      float au = (ku < 8) ? permx16(act[ku]) : act[ku & 7];
      frag[j][v] = hi ? au : al;
    }
  }
}

__global__ __launch_bounds__(256) void mlp_wmma_kernel(
    const float* __restrict__ x, const float* __restrict__ W0,
    const float* __restrict__ b0, const float* __restrict__ s0,
    const float* __restrict__ t0, const float* __restrict__ W1,
    const float* __restrict__ b1, const float* __restrict__ s1,
    const float* __restrict__ t1, const float* __restrict__ W2,
    const float* __restrict__ b2, const float* __restrict__ s2,
    const float* __restrict__ t2, const float* __restrict__ W3,
    const float* __restrict__ b3, const float* __restrict__ s3,
    const float* __restrict__ t3, const float* __restrict__ W4,
    const float* __restrict__ b4, const float* __restrict__ s4,
    const float* __restrict__ t4, const float* __restrict__ W5,
    const float* __restrict__ b5, const float* __restrict__ s5,
    const float* __restrict__ t5, const float* __restrict__ Wh,
    const float* __restrict__ bh, float* __restrict__ out, int N, int ntiles) {
  const int lane = threadIdx.x & 31;
  const int lane16 = lane & 15;
  const int hi = lane >> 4;
  const int wid = blockIdx.x * (blockDim.x >> 5) + (threadIdx.x >> 5);
  const int nw = gridDim.x * (blockDim.x >> 5);

  // One-time per-wave: weight A-fragments (zero-padded, clipped) and per-lane
  // (bias, s, t) selects. Amortized over the grid-stride tile loop.
  v2f wA0[2], wA1[4], wA2[4], wA3[3], wA4[2], wA5[1];
  load_wfrag<16, 8, false>(W0, wA0, lane16, hi);
  load_wfrag<16, 16, true>(W1, wA1, lane16, hi);
  load_wfrag<12, 16, true>(W2, wA2, lane16, hi);
  load_wfrag<8, 12, true>(W3, wA3, lane16, hi);
  load_wfrag<4, 8, true>(W4, wA4, lane16, hi);
  load_wfrag<4, 4, true>(W5, wA5, lane16, hi);

  float bs0[8], ss0[8], ts0[8], bs1[8], ss1[8], ts1[8];
  float bs2[8], ss2[8], ts2[8], bs3[8], ss3[8], ts3[8];
  float bs4[8], ss4[8], ts4[8], bs5[8], ss5[8], ts5[8];
  load_bst<16, false>(b0, s0, t0, bs0, ss0, ts0, hi);
  load_bst<16, true>(b1, s1, t1, bs1, ss1, ts1, hi);
  load_bst<12, true>(b2, s2, t2, bs2, ss2, ts2, hi);
  load_bst<8, true>(b3, s3, t3, bs3, ss3, ts3, hi);
  load_bst<4, true>(b4, s4, t4, bs4, ss4, ts4, hi);
  load_bst<4, true>(b5, s5, t5, bs5, ss5, ts5, hi);

  const float wh0 = Wh[0], wh1 = Wh[1], wh2 = Wh[2], wh3 = Wh[3];
  const float bhv = bh[0];

  for (int tile = wid; tile < ntiles; tile += nw) {
    const int row = tile * 16 + lane16;
    const bool rowok = (row < N);
    v2f bf[4];
    float act[8];

    // Layer 0 input: B element (vgpr v, slice j) = x[row][4*j + 2*hi + v]
    v2f z;
    z[0] = 0.0f;
    z[1] = 0.0f;
    bf[0] = z;
    bf[1] = z;
    if (rowok) {
      const float* xr = x + (size_t)row * 8 + 2 * hi;
      bf[0] = *(const v2f*)(xr + 0);
      bf[1] = *(const v2f*)(xr + 4);
    }
    // Branchless prefetch of the next tile's input row (near scope).
    {
      int ptile = tile + nw;
      if (ptile >= ntiles) ptile = tile;  // clamped, always-valid address
      __builtin_prefetch(x + (size_t)(ptile * 16 + lane16) * 8 + 2 * hi, 0, 3);
    }

    v8f c = zero8();
    c = wmma4(wA0[0], bf[0], c);
    c = wmma4(wA0[1], bf[1], c);
    activate<16>(c, bs0, ss0, ts0, act);

    make_bfrag<16>(act, bf, hi);                       // L1: 16 -> 16
    c = zero8();
#pragma unroll
    for (int j = 0; j < 4; ++j) c = wmma4(wA1[j], bf[j], c);
    activate<16>(c, bs1, ss1, ts1, act);

    make_bfrag<16>(act, bf, hi);                       // L2: 16 -> 12
    c = zero8();
#pragma unroll
    for (int j = 0; j < 4; ++j) c = wmma4(wA2[j], bf[j], c);
    activate<12>(c, bs2, ss2, ts2, act);

    make_bfrag<12>(act, bf, hi);                       // L3: 12 -> 8
    c = zero8();
#pragma unroll
    for (int j = 0; j < 3; ++j) c = wmma4(wA3[j], bf[j], c);
    activate<8>(c, bs3, ss3, ts3, act);

    make_bfrag<8>(act, bf, hi);                        // L4: 8 -> 4
    c = zero8();
#pragma unroll
    for (int j = 0; j < 2; ++j) c = wmma4(wA4[j], bf[j], c);
    activate<4>(c, bs4, ss4, ts4, act);

    make_bfrag<4>(act, bf, hi);                        // L5: 4 -> 4
    c = zero8();
    c = wmma4(wA5[0], bf[0], c);
    activate<4>(c, bs5, ss5, ts5, act);

    // Head (1x4) + sigmoid: valid rows live in the lower half-wave.
    float h = act[0] * wh0 + act[1] * wh1 + act[2] * wh2 + act[3] * wh3 + bhv;
    float y = fast_sigmoid(h);
    if (!hi && rowok) out[row] = y;  // 16 consecutive floats -> coalesced 64B
  }
}

extern "C" void kernel_launch(void* const* d_in, const int* in_sizes, int n_in,
                              void* d_out, int out_size, void* d_ws,
                              size_t ws_size, hipStream_t stream) {
  const float* x = (const float*)d_in[0];
  const float* W0 = (const float*)d_in[1];
  const float* b0 = (const float*)d_in[2];
  const float* s0 = (const float*)d_in[3];
  const float* t0 = (const float*)d_in[4];
  const float* W1 = (const float*)d_in[5];
  const float* b1 = (const float*)d_in[6];
  const float* s1 = (const float*)d_in[7];
  const float* t1 = (const float*)d_in[8];
  const float* W2 = (const float*)d_in[9];
  const float* b2 = (const float*)d_in[10];
  const float* s2 = (const float*)d_in[11];
  const float* t2 = (const float*)d_in[12];
  const float* W3 = (const float*)d_in[13];
  const float* b3 = (const float*)d_in[14];
  const float* s3 = (const float*)d_in[15];
  const float* t3 = (const float*)d_in[16];
  const float* W4 = (const float*)d_in[17];
  const float* b4 = (const float*)d_in[18];
  const float* s4 = (const float*)d_in[19];
  const float* t4 = (const float*)d_in[20];
  const float* W5 = (const float*)d_in[21];
  const float* b5 = (const float*)d_in[22];
  const float* s5 = (const float*)d_in[23];
  const float* t5 = (const float*)d_in[24];
  const float* Wh = (const float*)d_in[25];
  const float* bh = (const float*)d_in[26];

  const int N = in_sizes[0] / 8;
  const int ntiles = (N + 15) / 16;
  int blocks = (ntiles + 7) / 8;  // ~8 tiles per block, i.e. ~1 per wave (8 waves/block)
  if (blocks > 1024) blocks = 1024;
  if (blocks < 1) blocks = 1;

  mlp_wmma_kernel<<<blocks, 256, 0, stream>>>(
      x, W0, b0, s0, t0, W1, b1, s1, t1, W2, b2, s2, t2, W3, b3, s3, t3, W4, b4,
      s4, t4, W5, b5, s5, t5, Wh, bh, (float*)d_out, N, ntiles);
}